// RelationalGraphAttentionConv_21766894256814
// MI455X (gfx1250) — compile-verified
//
#include <hip/hip_runtime.h>
#include <hip/hip_bf16.h>

#define RP1 9
#define NHEAD 8
#define DIMS 128
#define EPSV 1e-10f
#define NEG_SLOPE 0.2f

typedef __attribute__((ext_vector_type(2))) float v2f;
typedef __attribute__((ext_vector_type(8))) float v8f;

// ---------------------------------------------------------------------------
// Kernel 0: initialize workspace accumulators (ws is poisoned by the harness)
// ---------------------------------------------------------------------------
__global__ void rgat_init_kernel(float* __restrict__ wmax, float* __restrict__ normsum,
                                 float* __restrict__ cnt, float* __restrict__ upd, int N) {
    int t = blockIdx.x * blockDim.x + threadIdx.x;
    if (t < N * DIMS) upd[t] = 0.0f;
    if (t < N * NHEAD) {
        normsum[t] = 0.0f;
        ((unsigned int*)wmax)[t] = 0xFF800000u;  // -inf
    }
    if (t < N) cnt[t] = 0.0f;
}

// ---------------------------------------------------------------------------
// Kernel 1: hidden[r] = x @ W_tau[r]^T via V_WMMA_F32_16X16X4_F32 (fp32 exact)
// Block = 128 threads (4 waves). Each block: 16 rows x 128 outs for one (r, row tile).
// Each wave owns two 16x16 output tiles; K-loop of 32 x (K=4) WMMA steps.
// A-frag:  lane(16*half+j) holds x[row0+j][k+2*half+{0,1}]      (float2 load)
// B-frag:  lane(16*half+j) holds W[o+j][k+2*half+{0,1}]         (float2 load)
// C/D:     vgpr v, lane(16*half+j) = C[v+8*half][j]
// ---------------------------------------------------------------------------
__global__ __launch_bounds__(128)
void rgat_gemm_hidden_kernel(const float* __restrict__ x, const float* __restrict__ W,
                             float* __restrict__ hidden, int N) {
    const int r    = blockIdx.y;
    const int row0 = blockIdx.x * 16;
    const int wave = threadIdx.x >> 5;    // 0..3
    const int lane = threadIdx.x & 31;
    const int hf   = lane >> 4;           // half: 0 or 1
    const int j    = lane & 15;

    const int arow = (row0 + j < N) ? (row0 + j) : (N - 1);  // clamp tail
    const float* __restrict__ Wr = W + (size_t)r * DIMS * DIMS;
    const int o0 = wave * 32;

    const float* __restrict__ ap  = x  + (size_t)arow * DIMS + 2 * hf;
    const float* __restrict__ bp0 = Wr + (size_t)(o0 + j) * DIMS + 2 * hf;
    const float* __restrict__ bp1 = Wr + (size_t)(o0 + 16 + j) * DIMS + 2 * hf;

    v8f acc0 = {};
    v8f acc1 = {};
#pragma unroll 4
    for (int k = 0; k < DIMS; k += 4) {
        v2f a  = *(const v2f*)(ap  + k);
        v2f b0 = *(const v2f*)(bp0 + k);
        v2f b1 = *(const v2f*)(bp1 + k);
        acc0 = __builtin_amdgcn_wmma_f32_16x16x4_f32(false, a, false, b0,
                                                     (short)0, acc0, false, false);
        acc1 = __builtin_amdgcn_wmma_f32_16x16x4_f32(false, a, false, b1,
                                                     (short)0, acc1, false, false);
    }

    float* __restrict__ hbase = hidden + ((size_t)r * N + row0) * DIMS;
#pragma unroll
    for (int v = 0; v < 8; ++v) {
        int m = v + 8 * hf;
        if (row0 + m < N) {
            hbase[(size_t)m * DIMS + (o0 + j)]      = acc0[v];
            hbase[(size_t)m * DIMS + (o0 + 16 + j)] = acc1[v];
        }
    }
}

// ---------------------------------------------------------------------------
// Kernel 2: per-(relation, node, head) query projections.
// a_in[r,n,h]  = sum_k query[r,h,2k]   * hidden[r,n,h*16+k]
// a_out[r,n,h] = sum_k query[r,h,2k+1] * hidden[r,n,h*16+k]
// (collapses the per-edge 512B gathers down to 32B gathers)
// ---------------------------------------------------------------------------
__global__ void rgat_aproj_kernel(const float* __restrict__ hidden,
                                  const float* __restrict__ query,
                                  float* __restrict__ a_in, float* __restrict__ a_out,
                                  int N) {
    size_t t = (size_t)blockIdx.x * blockDim.x + threadIdx.x;
    size_t total = (size_t)RP1 * N * NHEAD;
    if (t >= total) return;
    int h = (int)(t % NHEAD);
    size_t rn = t / NHEAD;          // r*N + n
    int r = (int)(rn / N);
    const float* __restrict__ hv = hidden + rn * DIMS + h * 16;
    const float* __restrict__ q  = query + ((size_t)r * NHEAD + h) * 32;
    float si = 0.0f, so = 0.0f;
#pragma unroll
    for (int k = 0; k < 16; ++k) {
        float hk = hv[k];
        si += q[2 * k]     * hk;
        so += q[2 * k + 1] * hk;
    }
    a_in[t]  = si;
    a_out[t] = so;
}

// ---------------------------------------------------------------------------
// Float atomic-max via the ordered-int-bits trick (wmax preinit to -inf bits)
// ---------------------------------------------------------------------------
__device__ __forceinline__ void atomicMaxF32(float* addr, float v) {
    if (v >= 0.0f) atomicMax((int*)addr, __float_as_int(v));
    else           atomicMin((unsigned int*)addr, (unsigned int)__float_as_int(v));
}

__device__ __forceinline__ void decode_edge(const int* __restrict__ el, int e, int E,
                                            int& src, int& dst, int& rel) {
    if (e < E) {
        src = el[3 * e]; dst = el[3 * e + 1]; rel = el[3 * e + 2];
    } else {
        src = dst = e - E; rel = RP1 - 1;  // appended self-loop
    }
}

// ---------------------------------------------------------------------------
// Kernel 3: segment max of leaky_relu(w) over dst, plus in-degree counts.
// ---------------------------------------------------------------------------
__global__ void rgat_edge_max_kernel(const int* __restrict__ el,
                                     const float* __restrict__ a_in,
                                     const float* __restrict__ a_out,
                                     float* __restrict__ wmax, float* __restrict__ cnt,
                                     int E, int N) {
    int e = blockIdx.x * blockDim.x + threadIdx.x;
    if (e >= E + N) return;
    int src, dst, rel;
    decode_edge(el, e, E, src, dst, rel);
    const float* __restrict__ ai = a_in  + ((size_t)rel * N + src) * NHEAD;
    const float* __restrict__ ao = a_out + ((size_t)rel * N + dst) * NHEAD;
#pragma unroll
    for (int h = 0; h < NHEAD; ++h) {
        float w = ai[h] + ao[h];
        w = (w > 0.0f) ? w : NEG_SLOPE * w;
        atomicMaxF32(&wmax[(size_t)dst * NHEAD + h], w);
    }
    atomicAdd(&cnt[dst], 1.0f);
}

// ---------------------------------------------------------------------------
// Kernel 4: segment sum of att = exp(w - wmax[dst]) * ew over dst.
// ---------------------------------------------------------------------------
__global__ void rgat_edge_norm_kernel(const int* __restrict__ el,
                                      const float* __restrict__ ew,
                                      const float* __restrict__ a_in,
                                      const float* __restrict__ a_out,
                                      const float* __restrict__ wmax,
                                      float* __restrict__ normsum,
                                      int E, int N) {
    int e = blockIdx.x * blockDim.x + threadIdx.x;
    if (e >= E + N) return;
    int src, dst, rel;
    decode_edge(el, e, E, src, dst, rel);
    float weight = (e < E) ? ew[e] : 1.0f;
    const float* __restrict__ ai = a_in  + ((size_t)rel * N + src) * NHEAD;
    const float* __restrict__ ao = a_out + ((size_t)rel * N + dst) * NHEAD;
    const float* __restrict__ wm = wmax + (size_t)dst * NHEAD;
#pragma unroll
    for (int h = 0; h < NHEAD; ++h) {
        float w = ai[h] + ao[h];
        w = (w > 0.0f) ? w : NEG_SLOPE * w;
        float att = __expf(w - wm[h]) * weight;
        atomicAdd(&normsum[(size_t)dst * NHEAD + h], att);
    }
}

// ---------------------------------------------------------------------------
// Kernel 5: one wave (32 lanes) per edge. Lane l covers outputs [4l, 4l+4);
// its head is l>>2. coeff = att / (normsum/cnt + EPS); scatter-add the message.
// ---------------------------------------------------------------------------
__global__ void rgat_edge_msg_kernel(const int* __restrict__ el,
                                     const float* __restrict__ ew,
                                     const float* __restrict__ a_in,
                                     const float* __restrict__ a_out,
                                     const float* __restrict__ wmax,
                                     const float* __restrict__ normsum,
                                     const float* __restrict__ cnt,
                                     const float* __restrict__ hidden,
                                     float* __restrict__ upd,
                                     int E, int N) {
    int gw = (blockIdx.x * blockDim.x + threadIdx.x) >> 5;  // global wave id = edge
    int lane = threadIdx.x & 31;
    if (gw >= E + N) return;
    int src, dst, rel;
    decode_edge(el, gw, E, src, dst, rel);
    int h = lane >> 2;
    float w = a_in[((size_t)rel * N + src) * NHEAD + h] +
              a_out[((size_t)rel * N + dst) * NHEAD + h];
    w = (w > 0.0f) ? w : NEG_SLOPE * w;
    float weight = (gw < E) ? ew[gw] : 1.0f;
    float att = __expf(w - wmax[(size_t)dst * NHEAD + h]) * weight;
    float c = cnt[dst];
    float coeff = att / (normsum[(size_t)dst * NHEAD + h] / c + EPSV);

    const float4 hv = *(const float4*)(hidden + ((size_t)rel * N + src) * DIMS + lane * 4);
    float* __restrict__ up = upd + (size_t)dst * DIMS + lane * 4;
    atomicAdd(up + 0, coeff * hv.x);
    atomicAdd(up + 1, coeff * hv.y);
    atomicAdd(up + 2, coeff * hv.z);
    atomicAdd(up + 3, coeff * hv.w);
}

// ---------------------------------------------------------------------------
// Kernel 6: out = relu(upd / cnt)
// ---------------------------------------------------------------------------
__global__ void rgat_finalize_kernel(const float* __restrict__ upd,
                                     const float* __restrict__ cnt,
                                     float* __restrict__ out, int N) {
    int t = blockIdx.x * blockDim.x + threadIdx.x;
    if (t >= N * DIMS) return;
    float v = upd[t] / cnt[t / DIMS];
    out[t] = (v > 0.0f) ? v : 0.0f;
}

// ---------------------------------------------------------------------------
extern "C" void kernel_launch(void* const* d_in, const int* in_sizes, int n_in,
                              void* d_out, int out_size, void* d_ws, size_t ws_size,
                              hipStream_t stream) {
    const float* x    = (const float*)d_in[0];
    const int*   el   = (const int*)d_in[1];
    const float* ew   = (const float*)d_in[2];
    const float* Wtau = (const float*)d_in[3];
    const float* qry  = (const float*)d_in[4];
    float* out = (float*)d_out;

    const int N = in_sizes[0] / DIMS;
    const int E = in_sizes[1] / 3;
    const int Etot = E + N;

    // Workspace layout (floats)
    float* ws      = (float*)d_ws;
    float* hidden  = ws;                                   // RP1*N*128
    float* a_in    = hidden + (size_t)RP1 * N * DIMS;      // RP1*N*8
    float* a_out   = a_in + (size_t)RP1 * N * NHEAD;       // RP1*N*8
    float* wmax    = a_out + (size_t)RP1 * N * NHEAD;      // N*8
    float* normsum = wmax + (size_t)N * NHEAD;             // N*8
    float* cnt     = normsum + (size_t)N * NHEAD;          // N
    float* upd     = cnt + N;                              // N*128

    // 0) init accumulators
    {
        int total = N * DIMS;
        rgat_init_kernel<<<(total + 255) / 256, 256, 0, stream>>>(wmax, normsum, cnt, upd, N);
    }
    // 1) WMMA GEMM: hidden[r] = x @ W_tau[r]^T
    {
        dim3 grid((N + 15) / 16, RP1);
        rgat_gemm_hidden_kernel<<<grid, 128, 0, stream>>>(x, Wtau, hidden, N);
    }
    // 2) per-(r,n,h) query projections
    {
        size_t total = (size_t)RP1 * N * NHEAD;
        rgat_aproj_kernel<<<(unsigned)((total + 255) / 256), 256, 0, stream>>>(
            hidden, qry, a_in, a_out, N);
    }
    // 3) segment max + counts
    rgat_edge_max_kernel<<<(Etot + 255) / 256, 256, 0, stream>>>(el, a_in, a_out, wmax, cnt, E, N);
    // 4) normalization sums
    rgat_edge_norm_kernel<<<(Etot + 255) / 256, 256, 0, stream>>>(el, ew, a_in, a_out, wmax,
                                                                  normsum, E, N);
    // 5) message scatter (wave per edge)
    {
        size_t threads = (size_t)Etot * 32;
        rgat_edge_msg_kernel<<<(unsigned)((threads + 255) / 256), 256, 0, stream>>>(
            el, ew, a_in, a_out, wmax, normsum, cnt, hidden, upd, E, N);
    }
    // 6) finalize
    rgat_finalize_kernel<<<(N * DIMS + 255) / 256, 256, 0, stream>>>(upd, cnt, out, N);
}